// TensorDense_13821204759104
// MI455X (gfx1250) — compile-verified
//
#include <hip/hip_runtime.h>

// ---------------------------------------------------------------------------
// e3nn-style equivariant block for MI455X (gfx1250, wave32, WMMA).
// GEMMs -> V_WMMA_F32_16X16X4_F32 (f32-faithful matrix pipe), with
// double-buffered GLOBAL_LOAD_ASYNC_TO_LDS staging (ASYNCcnt) when available.
//   stage1: 8x  [32768,512]x[512,256]   (a0,b0,a1_xyz,b1_xyz)
//   stage2: elementwise Clebsch-Gordan tensor product
//   stage3: 4x  [32768,{512,768}]x[.,512] + 5x [32768,256]x[256,256]
// ---------------------------------------------------------------------------

typedef float v2f __attribute__((ext_vector_type(2)));
typedef float v8f __attribute__((ext_vector_type(8)));
typedef int   v4i __attribute__((ext_vector_type(4)));

#define BM 64
#define BN 64
#define BK 16
#define LDAS 20   // A-tile LDS stride: rows 80B apart -> 16B aligned for b128
                  // writes; 20*m mod 64 distinct for m=0..15 -> conflict-free.

#ifndef __has_builtin
#define __has_builtin(x) 0
#endif
#if __has_builtin(__builtin_amdgcn_global_load_async_to_lds_b32) && \
    __has_builtin(__builtin_amdgcn_global_load_async_to_lds_b128)
#define HAVE_ASYNC_LDS 1
#else
#define HAVE_ASYNC_LDS 0
#endif

// ---- global -> LDS copy primitives ----------------------------------------
#if HAVE_ASYNC_LDS
typedef __attribute__((address_space(1))) int  gi_t;   // global int
typedef __attribute__((address_space(3))) int  li_t;   // LDS int
typedef __attribute__((address_space(1))) v4i  gv4_t;  // global int4
typedef __attribute__((address_space(3))) v4i  lv4_t;  // LDS int4

__device__ __forceinline__ void cp_b32(const float* __restrict__ g, float* l) {
    __builtin_amdgcn_global_load_async_to_lds_b32((gi_t*)g, (li_t*)l, 0, 0);
}
__device__ __forceinline__ void cp_b128(const float* __restrict__ g, float* l) {
    __builtin_amdgcn_global_load_async_to_lds_b128((gv4_t*)g, (lv4_t*)l, 0, 0);
}
#else
struct alignas(16) f4 { float x, y, z, w; };
__device__ __forceinline__ void cp_b32(const float* __restrict__ g, float* l) {
    *l = *g;
}
__device__ __forceinline__ void cp_b128(const float* __restrict__ g, float* l) {
    *(f4*)l = *(const f4*)g;
}
#endif

// Wait for outstanding async-LDS traffic, then block barrier.
__device__ __forceinline__ void lds_fence() {
#if HAVE_ASYNC_LDS
#if __has_builtin(__builtin_amdgcn_s_wait_asynccnt)
    __builtin_amdgcn_s_wait_asynccnt(0);
#else
    asm volatile("s_wait_asynccnt 0x0" ::: "memory");
#endif
#endif
    __syncthreads();
}

// ---- stage one BMxBK A tile + BKxBN B tile into LDS (128 threads) ---------
__device__ __forceinline__ void load_stage(
    const float* __restrict__ A, int lda, int esA,
    const float* __restrict__ W, int N,
    int m0, int n0, int kt, int tid,
    float* __restrict__ as, float* __restrict__ bs)
{
    if (esA == 1) {
        // contiguous rows: 16B chunks, 256 chunks total
        #pragma unroll
        for (int i = 0; i < 2; ++i) {
            int idx = tid + i * 128;
            int mm  = idx >> 2;            // 4 chunks per row
            int kc  = (idx & 3) * 4;
            cp_b128(&A[(size_t)(m0 + mm) * lda + kt + kc], &as[mm * LDAS + kc]);
        }
    } else {
        // strided (x1 component planes, esA=3): per-lane scalar gather
        #pragma unroll
        for (int i = 0; i < 8; ++i) {
            int idx = tid + i * 128;
            int mm  = idx >> 4;
            int kk  = idx & 15;
            cp_b32(&A[(size_t)(m0 + mm) * lda + (size_t)(kt + kk) * esA],
                   &as[mm * LDAS + kk]);
        }
    }
    // B tile: rows of 64 floats, 16 chunks each
    #pragma unroll
    for (int i = 0; i < 2; ++i) {
        int idx = tid + i * 128;
        int kk  = idx >> 4;
        int nc  = (idx & 15) * 4;
        cp_b128(&W[(size_t)(kt + kk) * N + n0 + nc], &bs[kk * BN + nc]);
    }
}

// ---- 4 WMMA K-steps (one LDS stage) for this wave's 2x2 fragment tile -----
__device__ __forceinline__ void compute_stage(
    const float* __restrict__ as, const float* __restrict__ bs,
    int wm, int wn, int hsel, int l16, v8f (&acc)[2][2])
{
    #pragma unroll
    for (int ks = 0; ks < BK; ks += 4) {
        const int kb = ks + hsel * 2;      // ISA 16x4 layout: half-wave K base
        v2f afr[2], bfr[2];
        #pragma unroll
        for (int t = 0; t < 2; ++t) {
            const int row = wm + t * 16 + l16;
            afr[t].x = as[row * LDAS + kb];
            afr[t].y = as[row * LDAS + kb + 1];
            const int col = wn + t * 16 + l16;
            bfr[t].x = bs[kb * BN + col];
            bfr[t].y = bs[(kb + 1) * BN + col];
        }
        #pragma unroll
        for (int tm = 0; tm < 2; ++tm)
            #pragma unroll
            for (int tn = 0; tn < 2; ++tn)
                acc[tm][tn] = __builtin_amdgcn_wmma_f32_16x16x4_f32(
                    false, afr[tm], false, bfr[tn],
                    (short)0, acc[tm][tn], false, false);
    }
}

// Generic tiled GEMM: C[m,n] = scale * sum_k A[m,k] * W[k,n]
//   A element (m,k) at A[m*lda + k*esA]; W row-major [K,N];
//   C element (m,n) at C[m*ldc + n*esC].
// 128 threads = 4 waves; block tile 64x64; wave tile 32x32 (2x2 fragments).
// Double-buffered LDS: next K-stage streams in (async) while WMMAs run.
__device__ __forceinline__ void gemm_tile(const float* __restrict__ A, int lda, int esA,
                                          const float* __restrict__ W, int N, int K,
                                          float* __restrict__ C, int ldc, int esC,
                                          float scale)
{
    __shared__ float As0[BM * LDAS], As1[BM * LDAS];
    __shared__ float Bs0[BK * BN],  Bs1[BK * BN];

    const int tid  = threadIdx.x;
    const int lane = tid & 31;
    const int wave = tid >> 5;            // 0..3
    const int wm   = (wave >> 1) * 32;
    const int wn   = (wave & 1) * 32;
    const int m0   = blockIdx.y * BM;
    const int n0   = blockIdx.x * BN;
    const int hsel = lane >> 4;           // lanes 0-15: K+0,1; lanes 16-31: K+2,3
    const int l16  = lane & 15;

    v8f acc[2][2] = {};

    load_stage(A, lda, esA, W, N, m0, n0, 0, tid, As0, Bs0);
    lds_fence();

    const int nt = K / BK;                // 32 / 48 / 16 -> always even
    for (int t = 0; t < nt; t += 2) {
        load_stage(A, lda, esA, W, N, m0, n0, (t + 1) * BK, tid, As1, Bs1);
        compute_stage(As0, Bs0, wm, wn, hsel, l16, acc);
        lds_fence();
        if (t + 2 < nt)
            load_stage(A, lda, esA, W, N, m0, n0, (t + 2) * BK, tid, As0, Bs0);
        compute_stage(As1, Bs1, wm, wn, hsel, l16, acc);
        lds_fence();
    }

    // epilogue: C/D layout = VGPR r: lanes0-15 M=r, lanes16-31 M=r+8
    #pragma unroll
    for (int tm = 0; tm < 2; ++tm)
        #pragma unroll
        for (int tn = 0; tn < 2; ++tn) {
            const int col = n0 + wn + tn * 16 + l16;
            #pragma unroll
            for (int r = 0; r < 8; ++r) {
                const int row = m0 + wm + tm * 16 + r + hsel * 8;
                C[(size_t)row * ldc + (size_t)col * esC] = acc[tm][tn][r] * scale;
            }
        }
}

// ---------------------------------------------------------------------------
// Stage 1: 8 GEMMs -> a0, b0, a1[3 planes], b1[3 planes]   (all [B,256])
// ---------------------------------------------------------------------------
__global__ __launch_bounds__(128) void stage1_gemm(
    const float* __restrict__ x0, const float* __restrict__ x1,
    const float* __restrict__ W1_0, const float* __restrict__ W1_1,
    const float* __restrict__ W2_0, const float* __restrict__ W2_1,
    float* __restrict__ a0, float* __restrict__ b0,
    float* __restrict__ a1, float* __restrict__ b1)
{
    const size_t P = (size_t)32768 * 256;
    const float s512 = 0.04419417382415922f;  // 512^-0.5
    const int z = blockIdx.z;
    const float* A; int lda, esA; const float* W; float* C;
    if (z == 0)      { A = x0;            lda = 512;  esA = 1; W = W1_0; C = a0; }
    else if (z == 1) { A = x0;            lda = 512;  esA = 1; W = W2_0; C = b0; }
    else if (z <= 4) { int i = z - 2; A = x1 + i; lda = 1536; esA = 3; W = W1_1; C = a1 + (size_t)i * P; }
    else             { int i = z - 5; A = x1 + i; lda = 1536; esA = 3; W = W2_1; C = b1 + (size_t)i * P; }
    gemm_tile(A, lda, esA, W, 256, 512, C, 256, 1, s512);
}

// ---------------------------------------------------------------------------
// Stage 2: channel-wise Clebsch-Gordan tensor product (elementwise).
// tp0:[B,512]  tp1:[3][B,768] (component-major)  tp2:[5][B,256]
// ---------------------------------------------------------------------------
__global__ __launch_bounds__(256) void stage2_tp(
    const float* __restrict__ a0p, const float* __restrict__ b0p,
    const float* __restrict__ a1p, const float* __restrict__ b1p,
    float* __restrict__ tp0, float* __restrict__ tp1, float* __restrict__ tp2)
{
    const size_t P = (size_t)32768 * 256;
    const size_t t = (size_t)blockIdx.x * 256 + threadIdx.x;  // b*256 + c
    const size_t b = t >> 8;
    const int    c = (int)(t & 255);

    const float A0 = a0p[t], B0 = b0p[t];
    const float ax = a1p[t], ay = a1p[t + P], az = a1p[t + 2 * P];
    const float bx = b1p[t], by = b1p[t + P], bz = b1p[t + 2 * P];

    const float inv3 = 0.5773502691896258f;   // 1/sqrt(3)
    const float inv2 = 0.7071067811865476f;   // 1/sqrt(2)
    const float inv6 = 0.4082482904638630f;   // 1/sqrt(6)

    tp0[b * 512 + c]       = A0 * B0;
    tp0[b * 512 + 256 + c] = inv3 * (ax * bx + ay * by + az * bz);

    const float cr[3]  = { ay * bz - az * by, az * bx - ax * bz, ax * by - ay * bx };
    const float a1v[3] = { ax, ay, az };
    const float b1v[3] = { bx, by, bz };
    const size_t Q = (size_t)32768 * 768;
    #pragma unroll
    for (int i = 0; i < 3; ++i) {
        float* p = tp1 + (size_t)i * Q + b * 768;
        p[c]       = A0 * b1v[i];
        p[256 + c] = B0 * a1v[i];
        p[512 + c] = inv2 * cr[i];
    }

    tp2[0 * P + t] = inv2 * (ax * by + ay * bx);                   // xy
    tp2[1 * P + t] = inv2 * (ay * bz + az * by);                   // yz
    tp2[2 * P + t] = inv6 * (2.0f * az * bz - ax * bx - ay * by);  // 3z^2-r^2
    tp2[3 * P + t] = inv2 * (ax * bz + az * bx);                   // xz
    tp2[4 * P + t] = inv2 * (ax * bx - ay * by);                   // x^2-y^2
}

// ---------------------------------------------------------------------------
// Stage 3a: y0 = tp0 @ W3_0  and  y1_i = tp1_i @ W3_1 (store at b*1536+o*3+i)
// ---------------------------------------------------------------------------
__global__ __launch_bounds__(128) void stage3_big(
    const float* __restrict__ tp0, const float* __restrict__ tp1,
    const float* __restrict__ W3_0, const float* __restrict__ W3_1,
    float* __restrict__ out)
{
    const size_t Bn = 32768;
    const int z = blockIdx.z;
    if (z == 0) {
        gemm_tile(tp0, 512, 1, W3_0, 512, 512, out, 512, 1,
                  0.04419417382415922f);                    // 512^-0.5
    } else {
        const int i = z - 1;
        gemm_tile(tp1 + (size_t)i * Bn * 768, 768, 1, W3_1, 512, 768,
                  out + Bn * 512 + i, 1536, 3,
                  0.03608439182435161f);                    // 768^-0.5
    }
}

// ---------------------------------------------------------------------------
// Stage 3b: y2_m = tp2_m @ W3_2, stored at b*1280 + o*5 + m
// ---------------------------------------------------------------------------
__global__ __launch_bounds__(128) void stage3_y2(
    const float* __restrict__ tp2, const float* __restrict__ W3_2,
    float* __restrict__ out)
{
    const size_t Bn = 32768;
    const int m = blockIdx.z;
    gemm_tile(tp2 + (size_t)m * Bn * 256, 256, 1, W3_2, 256, 256,
              out + Bn * 2048 + m, 1280, 5, 0.0625f);       // 256^-0.5
}

// ---------------------------------------------------------------------------
extern "C" void kernel_launch(void* const* d_in, const int* in_sizes, int n_in,
                              void* d_out, int out_size, void* d_ws, size_t ws_size,
                              hipStream_t stream)
{
    const float* x0   = (const float*)d_in[0];
    const float* x1   = (const float*)d_in[1];
    const float* W1_0 = (const float*)d_in[2];
    const float* W1_1 = (const float*)d_in[3];
    const float* W2_0 = (const float*)d_in[4];
    const float* W2_1 = (const float*)d_in[5];
    const float* W3_0 = (const float*)d_in[6];
    const float* W3_1 = (const float*)d_in[7];
    const float* W3_2 = (const float*)d_in[8];
    float* out = (float*)d_out;

    // Workspace, units of P = 32768*256 floats (32 MB):
    //   a0:1 b0:1 a1:3 b1:3 tp0:2 tp1:9 tp2:5  => 24P ~ 805 MB
    float* ws = (float*)d_ws;
    const size_t P = (size_t)32768 * 256;
    float* a0  = ws;
    float* b0  = ws + P;
    float* a1  = ws + 2 * P;
    float* b1  = ws + 5 * P;
    float* tp0 = ws + 8 * P;
    float* tp1 = ws + 10 * P;
    float* tp2 = ws + 19 * P;

    const dim3 blk(128);
    stage1_gemm<<<dim3(256 / BN, 32768 / BM, 8), blk, 0, stream>>>(
        x0, x1, W1_0, W1_1, W2_0, W2_1, a0, b0, a1, b1);
    stage2_tp<<<dim3((unsigned)(P / 256)), dim3(256), 0, stream>>>(
        a0, b0, a1, b1, tp0, tp1, tp2);
    stage3_big<<<dim3(512 / BN, 32768 / BM, 4), blk, 0, stream>>>(
        tp0, tp1, W3_0, W3_1, out);
    stage3_y2<<<dim3(256 / BN, 32768 / BM, 5), blk, 0, stream>>>(
        tp2, W3_2, out);
}